// LightGCN_32229434589220
// MI455X (gfx1250) — compile-verified
//
#include <hip/hip_runtime.h>
#include <math.h>

#define FDIM 128
#define N_LAYERS 3

typedef __attribute__((ext_vector_type(2))) float v2f;
typedef __attribute__((ext_vector_type(8))) float v8f;

// ---------------------------------------------------------------- utilities
__global__ __launch_bounds__(256) void k_zero(float* __restrict__ p, int n) {
  int i = blockIdx.x * blockDim.x + threadIdx.x;
  int stride = gridDim.x * blockDim.x;
  for (; i < n; i += stride) p[i] = 0.0f;
}

__global__ __launch_bounds__(256) void k_copy(float* __restrict__ dst,
                                              const float* __restrict__ src, int n) {
  int i = blockIdx.x * blockDim.x + threadIdx.x;
  int stride = gridDim.x * blockDim.x;
  for (; i < n; i += stride) dst[i] = src[i];
}

__global__ __launch_bounds__(256) void k_add(float* __restrict__ acc,
                                             const float* __restrict__ h, int n) {
  int i = blockIdx.x * blockDim.x + threadIdx.x;
  int stride = gridDim.x * blockDim.x;
  for (; i < n; i += stride) acc[i] += h[i];
}

// ---------------------------------------------------- degree / edge weights
__global__ __launch_bounds__(256) void k_degree(const int* __restrict__ row,
                                                const float* __restrict__ vals,
                                                float* __restrict__ deg, int e) {
  int i = blockIdx.x * blockDim.x + threadIdx.x;
  int stride = gridDim.x * blockDim.x;
  for (; i < e; i += stride) atomicAdd(&deg[row[i]], vals[i]);
}

__global__ __launch_bounds__(256) void k_dinv(float* __restrict__ deg, int n) {
  int i = blockIdx.x * blockDim.x + threadIdx.x;
  int stride = gridDim.x * blockDim.x;
  for (; i < n; i += stride) {
    float d = deg[i];
    d = (d > 0.0f) ? d : 1.0f;
    deg[i] = 1.0f / sqrtf(d);   // in place: deg becomes dinv
  }
}

__global__ __launch_bounds__(256) void k_nvals(const int* __restrict__ row,
                                               const int* __restrict__ col,
                                               const float* __restrict__ vals,
                                               const float* __restrict__ dinv,
                                               float* __restrict__ nv, int e) {
  int i = blockIdx.x * blockDim.x + threadIdx.x;
  int stride = gridDim.x * blockDim.x;
  for (; i < e; i += stride) nv[i] = vals[i] * dinv[row[i]] * dinv[col[i]];
}

// --------------------------------------------------- sparse propagation step
// One wave per edge: 32 lanes x float4 = the full 128-float feature row.
// Coalesced 512B gather of h[col], f32-atomic scatter-add into hnext[row].
__global__ __launch_bounds__(256) void k_prop(const int* __restrict__ row,
                                              const int* __restrict__ col,
                                              const float* __restrict__ nv,
                                              const float* __restrict__ h,
                                              float* __restrict__ hnext, int e) {
  const int lane = threadIdx.x & 31;
  int gw = (blockIdx.x * blockDim.x + threadIdx.x) >> 5;
  const int nw = (gridDim.x * blockDim.x) >> 5;
  for (int ei = gw; ei < e; ei += nw) {
    const int r = row[ei];
    const int c = col[ei];
    const float w = nv[ei];
    const float4 v = *((const float4*)(h + c * FDIM) + lane);
    float* dst = hnext + r * FDIM + lane * 4;
    atomicAdd(dst + 0, w * v.x);
    atomicAdd(dst + 1, w * v.y);
    atomicAdd(dst + 2, w * v.z);
    atomicAdd(dst + 3, w * v.w);
  }
}

// -------------------------------- WMMA projection + bias + L2-normalize + fp16
// Block = 16 nodes x 128 out cols, 8 waves; wave w owns cols [16w, 16w+16).
// D(16x16) accumulated by 32 chained v_wmma_f32_16x16x4_f32 over K=128.
__global__ __launch_bounds__(256) void k_out_gemm(const float* __restrict__ acc,
                                                  const float* __restrict__ Wm,
                                                  const float* __restrict__ bias,
                                                  _Float16* __restrict__ out, int nn) {
  __shared__ float As[16][FDIM + 2];   // stride 130: per-lane A reads hit 16 banks
  __shared__ float norm2[16];

  const int t = threadIdx.x;
  const int lane = t & 31;
  const int wave = t >> 5;
  const int m = lane & 15;            // matrix row (A) / col (B,D) within tile
  const int kh = (lane >> 4) << 1;    // K sub-offset: 0 (lanes 0-15) or 2 (16-31)
  const int rowbase = blockIdx.x * 16;
  const int ncol = wave * 16 + m;

  // Stage A tile (16x128), folding the /(NUM_LAYERS+1) mean into the load.
  {
    int f = t * 2;                    // 512 float4 total, 2 per thread
#pragma unroll
    for (int j = 0; j < 2; ++j, ++f) {
      const int r = f >> 5;           // 32 float4 per row
      const int cc = (f & 31) << 2;
      float4 v = make_float4(0.f, 0.f, 0.f, 0.f);
      if (rowbase + r < nn)
        v = *(const float4*)(acc + (rowbase + r) * FDIM + cc);
      As[r][cc + 0] = v.x * 0.25f;
      As[r][cc + 1] = v.y * 0.25f;
      As[r][cc + 2] = v.z * 0.25f;
      As[r][cc + 3] = v.w * 0.25f;
    }
  }
  if (t < 16) norm2[t] = 0.0f;
  __syncthreads();

  const float* bcol = Wm + kh * FDIM + ncol;
  v8f c = {0.f, 0.f, 0.f, 0.f, 0.f, 0.f, 0.f, 0.f};
#pragma unroll
  for (int k = 0; k < FDIM; k += 4) {
    v2f a;
    a.x = As[m][k + kh];
    a.y = As[m][k + kh + 1];
    v2f b;
    b.x = bcol[k * FDIM];
    b.y = bcol[(k + 1) * FDIM];
    c = __builtin_amdgcn_wmma_f32_16x16x4_f32(false, a, false, b,
                                              (short)0, c, false, false);
  }

  // D layout: VGPR v holds row (rb + v), col ncol.
  const float bv = bias[ncol];
  const int rb = (lane >> 4) << 3;    // 0 or 8
  float vv[8];
#pragma unroll
  for (int v = 0; v < 8; ++v) {
    vv[v] = c[v] + bv;
    atomicAdd(&norm2[rb + v], vv[v] * vv[v]);   // ds_add_f32 row reduction
  }
  __syncthreads();

#pragma unroll
  for (int v = 0; v < 8; ++v) {
    const int r = rowbase + rb + v;
    if (r < nn) {
      const float nrm = sqrtf(norm2[rb + v]);
      const float s = 0.1f / fmaxf(nrm, 1e-12f);
      out[r * FDIM + ncol] = (_Float16)(vv[v] * s);
    }
  }
}

// ---------------------------------------------------------------- launcher
extern "C" void kernel_launch(void* const* d_in, const int* in_sizes, int n_in,
                              void* d_out, int out_size, void* d_ws, size_t ws_size,
                              hipStream_t stream) {
  const int* row = (const int*)d_in[0];
  const int* col = (const int*)d_in[1];
  const float* vals = (const float*)d_in[2];
  const float* x = (const float*)d_in[3];
  const float* Wm = (const float*)d_in[4];
  const float* bias = (const float*)d_in[5];
  _Float16* out = (_Float16*)d_out;

  const int E = in_sizes[0];
  const int Nn = in_sizes[3] / FDIM;
  const int NF = Nn * FDIM;

  // Workspace layout (floats): deg[N] | nvals[E] | acc[N*F] | hA[N*F] | hB[N*F]
  float* ws = (float*)d_ws;
  float* deg = ws;
  float* nv = deg + Nn;
  float* acc = nv + E;
  float* hA = acc + NF;
  float* hB = hA + NF;

  const int B = 256;
  int gN = (Nn + B - 1) / B;
  int gE = (E + B - 1) / B;   if (gE > 16384) gE = 16384;
  int gNF = (NF + B - 1) / B; if (gNF > 16384) gNF = 16384;

  // Degree + symmetric normalization weights
  k_zero<<<gN, B, 0, stream>>>(deg, Nn);
  k_degree<<<gE, B, 0, stream>>>(row, vals, deg, E);
  k_dinv<<<gN, B, 0, stream>>>(deg, Nn);
  k_nvals<<<gE, B, 0, stream>>>(row, col, vals, deg, nv, E);

  // acc = x; then 3 propagation layers
  k_copy<<<gNF, B, 0, stream>>>(acc, x, NF);
  const float* hprev = x;
  float* bufs[2] = {hA, hB};
  for (int l = 0; l < N_LAYERS; ++l) {
    float* hnext = bufs[l & 1];
    k_zero<<<gNF, B, 0, stream>>>(hnext, NF);
    k_prop<<<8192, B, 0, stream>>>(row, col, nv, hprev, hnext, E);
    k_add<<<gNF, B, 0, stream>>>(acc, hnext, NF);
    hprev = hnext;
  }

  // Dense projection (WMMA f32) + bias + L2 normalize + fp16 store
  const int gOut = (Nn + 15) / 16;
  k_out_gemm<<<gOut, B, 0, stream>>>(acc, Wm, bias, out, Nn);
}